// DecompGrid_24910810317371
// MI455X (gfx1250) — compile-verified
//
#include <hip/hip_runtime.h>
#include <cstdint>
#include <cstddef>

// ---------------------------------------------------------------------------
// DecompGrid sampler for MI455X (gfx1250, wave32).
//   Pass 1: transpose channel-major tables (16,V) -> channel-last (V,16) in d_ws
//           (NT loads so the streaming read does not evict the transposed
//            tables, which we want resident in the 192MB L2).
//   Pass 2: 4 lanes per point; each lane owns 4 channels -> every corner gather
//           is one 16B global_load_b128, quad-coalesced into 64B requests.
//           Lines + x tile staged to LDS via global_load_async_to_lds_b128
//           (ASYNCcnt) and transposed in LDS. Output written non-temporally.
// ---------------------------------------------------------------------------

typedef __attribute__((ext_vector_type(4))) float v4f;

static constexpr int    NC  = 16;
static constexpr int    GSZ = 128;
static constexpr int    PSZ = 512;
static constexpr int    LSZ = 64;
static constexpr size_t V3  = (size_t)GSZ * GSZ * GSZ;  // 2,097,152 voxels
static constexpr size_t V2  = (size_t)PSZ * PSZ;        //   262,144 texels

// ---- CDNA5 async global->LDS (tracked by ASYNCcnt) ------------------------
__device__ __forceinline__ void async_ld16(uint32_t lds_byte_off, const void* gptr) {
  asm volatile("global_load_async_to_lds_b128 %0, %1, off"
               :: "v"(lds_byte_off), "v"(gptr)
               : "memory");
}
__device__ __forceinline__ void wait_async0() {
#if __has_builtin(__builtin_amdgcn_s_wait_asynccnt)
  __builtin_amdgcn_s_wait_asynccnt(0);
#else
  asm volatile("s_wait_asynccnt 0" ::: "memory");
#endif
}

// Reference's _axis_idx: pos = (c+1)*0.5*(size-1); clamp both corners.
__device__ __forceinline__ void axis_idx(float c, int size, int& i0, int& i1, float& w) {
  float pos = (c + 1.0f) * 0.5f * (float)(size - 1);
  float f   = floorf(pos);
  w  = pos - f;
  int i = (int)f;
  i0 = min(max(i, 0), size - 1);
  i1 = min(max(i + 1, 0), size - 1);
}

// 4 channels for this lane: channel-last => one b128; channel-major => 4 gathers.
template <bool CL>
__device__ __forceinline__ v4f gather4(const float* __restrict__ t, size_t vox,
                                       int q4, size_t V) {
  if constexpr (CL) {
    return *reinterpret_cast<const v4f*>(t + vox * NC + q4);
  } else {
    v4f r;
    r.x = t[(size_t)(q4 + 0) * V + vox];
    r.y = t[(size_t)(q4 + 1) * V + vox];
    r.z = t[(size_t)(q4 + 2) * V + vox];
    r.w = t[(size_t)(q4 + 3) * V + vox];
    return r;
  }
}

// ---- Pass 1: (16, V) channel-major -> (V, 16) channel-last ---------------
__global__ __launch_bounds__(256) void transpose_cl_k(const float* __restrict__ src,
                                                      float* __restrict__ dst, int V) {
  int v = blockIdx.x * 256 + threadIdx.x;
  if (v >= V) return;
  float t[NC];
#pragma unroll
  for (int c = 0; c < NC; ++c)
    t[c] = __builtin_nontemporal_load(src + (size_t)c * (size_t)V + v);  // NT: streaming
  v4f* o = reinterpret_cast<v4f*>(dst + (size_t)v * NC);
#pragma unroll
  for (int g = 0; g < 4; ++g) {
    v4f r = {t[4 * g], t[4 * g + 1], t[4 * g + 2], t[4 * g + 3]};
    o[g] = r;  // RT store: keep transposed tables hot in L2
  }
}

// ---- Pass 2: sample. 256 threads = 64 points, 4 lanes/point ---------------
template <bool CL>
__global__ __launch_bounds__(256) void sample_k(
    const float* __restrict__ x,
    const float* __restrict__ g3,
    const float* __restrict__ p0, const float* __restrict__ p1,
    const float* __restrict__ p2,
    const float* __restrict__ l0, const float* __restrict__ l1,
    const float* __restrict__ l2,
    float* __restrict__ out, int B) {
  const int tid  = threadIdx.x;
  const int q4   = (tid & 3) * 4;   // this lane's channel group
  const int lp   = tid >> 2;        // local point 0..63
  const int base = blockIdx.x * 64;
  const int p    = base + lp;

  __shared__ __attribute__((aligned(16))) float s_raw[3 * NC * LSZ];  // [k][c][i]
  __shared__ __attribute__((aligned(16))) float s_lin[3 * LSZ * NC];  // [k][i][c]
  __shared__ __attribute__((aligned(16))) float s_x[64 * 6];

  // Async-stage the three 4KB lines (each thread: 3 x 16B chunks).
  {
    const uint32_t rb = (uint32_t)(uintptr_t)&s_raw[0];
    const float* lines[3] = {l0, l1, l2};
#pragma unroll
    for (int k = 0; k < 3; ++k)
      async_ld16(rb + (uint32_t)(k * 4096 + tid * 16), lines[k] + tid * 4);
  }
  // Async-stage this block's x tile (<= 1536B).
  {
    const uint32_t xb  = (uint32_t)(uintptr_t)&s_x[0];
    const int cnt      = min(64, B - base);
    const int nfl      = cnt * 6;
    const int nchunk   = nfl >> 2;
    if (tid < nchunk)
      async_ld16(xb + (uint32_t)(tid * 16), x + (size_t)base * 6 + (size_t)tid * 4);
    const int rem = nfl & 3;
    if (tid < rem)
      s_x[nchunk * 4 + tid] = x[(size_t)base * 6 + (size_t)(nchunk * 4 + tid)];
  }
  wait_async0();
  __syncthreads();

  // LDS transpose of lines: [k][c][i] -> [k][i][c] (channel-last, b128-friendly).
#pragma unroll
  for (int j = 0; j < 12; ++j) {
    int t = tid + j * 256;
    int k = t >> 10, r = t & 1023, c = r >> 6, i = r & 63;
    s_lin[k * 1024 + i * NC + c] = s_raw[t];
  }
  __syncthreads();

  if (p >= B) return;

  float xc[6];
#pragma unroll
  for (int i = 0; i < 6; ++i) xc[i] = s_x[lp * 6 + i];

  // ---- trilinear on 128^3 grid ----
  int x0, x1, y0, y1, z0, z1;
  float wx, wy, wz;
  axis_idx(xc[0], GSZ, x0, x1, wx);
  axis_idx(xc[1], GSZ, y0, y1, wy);
  axis_idx(xc[2], GSZ, z0, z1, wz);
  const float ax = 1.0f - wx, ay = 1.0f - wy, az = 1.0f - wz;
  const size_t zy00 = (size_t)((z0 * GSZ + y0) * GSZ);
  const size_t zy01 = (size_t)((z0 * GSZ + y1) * GSZ);
  const size_t zy10 = (size_t)((z1 * GSZ + y0) * GSZ);
  const size_t zy11 = (size_t)((z1 * GSZ + y1) * GSZ);

  v4f sf;
  sf  = gather4<CL>(g3, zy00 + x0, q4, V3) * (az * ay * ax);
  sf += gather4<CL>(g3, zy00 + x1, q4, V3) * (az * ay * wx);
  sf += gather4<CL>(g3, zy01 + x0, q4, V3) * (az * wy * ax);
  sf += gather4<CL>(g3, zy01 + x1, q4, V3) * (az * wy * wx);
  sf += gather4<CL>(g3, zy10 + x0, q4, V3) * (wz * ay * ax);
  sf += gather4<CL>(g3, zy10 + x1, q4, V3) * (wz * ay * wx);
  sf += gather4<CL>(g3, zy11 + x0, q4, V3) * (wz * wy * ax);
  sf += gather4<CL>(g3, zy11 + x1, q4, V3) * (wz * wy * wx);

  // ---- three bilinear 512^2 planes, PLANE_DIMID = (0,1),(0,2),(1,2) ----
  {
    const float* planes[3] = {p0, p1, p2};
    const int pa[3] = {0, 0, 1};
    const int pb[3] = {1, 2, 2};
#pragma unroll
    for (int k = 0; k < 3; ++k) {
      int u0, u1, v0, v1;
      float wu, wv;
      axis_idx(xc[pa[k]], PSZ, u0, u1, wu);  // coords[:,0] -> W
      axis_idx(xc[pb[k]], PSZ, v0, v1, wv);  // coords[:,1] -> H
      const float au = 1.0f - wu, av = 1.0f - wv;
      const size_t r0 = (size_t)v0 * PSZ, r1 = (size_t)v1 * PSZ;
      v4f bl;
      bl  = gather4<CL>(planes[k], r0 + u0, q4, V2) * (av * au);
      bl += gather4<CL>(planes[k], r0 + u1, q4, V2) * (av * wu);
      bl += gather4<CL>(planes[k], r1 + u0, q4, V2) * (wv * au);
      bl += gather4<CL>(planes[k], r1 + u1, q4, V2) * (wv * wu);
      sf *= bl;
    }
  }

  // ---- three 64-entry lines from LDS (channel-last) ----
  v4f pf;
#pragma unroll
  for (int k = 0; k < 3; ++k) {
    float pn = xc[3 + k] * (float)(LSZ - 1);
    float f  = floorf(pn);
    float w  = pn - f;
    int i0 = (int)f;
    int i1 = min(i0 + 1, LSZ - 1);
    const float* lb = &s_lin[k * 1024];
    v4f a = *reinterpret_cast<const v4f*>(lb + i0 * NC + q4);
    v4f b = *reinterpret_cast<const v4f*>(lb + i1 * NC + q4);
    v4f s = a + w * (b - a);
    pf = (k == 0) ? s : pf * s;
  }

  // ---- output (B,32): [sf | pf], non-temporal (pure stream, spare the L2) --
  float* op = out + (size_t)p * 32;
  __builtin_nontemporal_store(sf, reinterpret_cast<v4f*>(op + q4));
  __builtin_nontemporal_store(pf, reinterpret_cast<v4f*>(op + 16 + q4));
}

// ---------------------------------------------------------------------------
extern "C" void kernel_launch(void* const* d_in, const int* in_sizes, int n_in,
                              void* d_out, int out_size, void* d_ws, size_t ws_size,
                              hipStream_t stream) {
  const float* x  = (const float*)d_in[0];
  const float* g3 = (const float*)d_in[1];
  const float* p0 = (const float*)d_in[2];
  const float* p1 = (const float*)d_in[3];
  const float* p2 = (const float*)d_in[4];
  const float* l0 = (const float*)d_in[5];
  const float* l1 = (const float*)d_in[6];
  const float* l2 = (const float*)d_in[7];
  float* out = (float*)d_out;
  const int B = in_sizes[0] / 6;

  const size_t need = (V3 + 3 * V2) * (size_t)NC * sizeof(float);  // 184,549,376 B
  const int nblk = (B + 63) / 64;

  if (d_ws != nullptr && ws_size >= need) {
    float* gt  = (float*)d_ws;
    float* p0t = gt + V3 * NC;
    float* p1t = p0t + V2 * NC;
    float* p2t = p1t + V2 * NC;
    transpose_cl_k<<<(int)(V3 / 256), 256, 0, stream>>>(g3, gt, (int)V3);
    transpose_cl_k<<<(int)(V2 / 256), 256, 0, stream>>>(p0, p0t, (int)V2);
    transpose_cl_k<<<(int)(V2 / 256), 256, 0, stream>>>(p1, p1t, (int)V2);
    transpose_cl_k<<<(int)(V2 / 256), 256, 0, stream>>>(p2, p2t, (int)V2);
    sample_k<true><<<nblk, 256, 0, stream>>>(x, gt, p0t, p1t, p2t, l0, l1, l2, out, B);
  } else {
    // Fallback: sample directly from channel-major tables (slower gathers).
    sample_k<false><<<nblk, 256, 0, stream>>>(x, g3, p0, p1, p2, l0, l1, l2, out, B);
  }
}